// SimModule_5394478924524
// MI455X (gfx1250) — compile-verified
//
#include <hip/hip_runtime.h>

#define GRID_DIM   128
#define T_SIG      400
#define ADC_SAMPLE 50
#define NB         200   // 10000 / 50 coarse bins
#define NBP        208   // padded to 13 * 16
#define NBLK       13

typedef __attribute__((ext_vector_type(2))) float v2f;
typedef __attribute__((ext_vector_type(8))) float v8f;

// ---------------------------------------------------------------------------
// Kernel 1a: init 128x128 coordinate -> pixel-index map to "miss" value U
// ---------------------------------------------------------------------------
__global__ void init_map_kernel(int* __restrict__ map, int U) {
    int i = blockIdx.x * blockDim.x + threadIdx.x;
    if (i < GRID_DIM * GRID_DIM) map[i] = U;
}

// ---------------------------------------------------------------------------
// Kernel 1b: scatter pool entries into the map (pool coords are unique)
// ---------------------------------------------------------------------------
__global__ void fill_map_kernel(const int* __restrict__ pool,
                                int* __restrict__ map, int U) {
    int u = blockIdx.x * blockDim.x + threadIdx.x;
    if (u < U) {
        int x = pool[2 * u + 0];
        int y = pool[2 * u + 1];
        map[x * GRID_DIM + y] = u;
    }
}

// ---------------------------------------------------------------------------
// Kernel 2: zero the (U+1) x NBP bin scratch
// ---------------------------------------------------------------------------
__global__ void zero_bins_kernel(float* __restrict__ bins, long n) {
    long i = (long)blockIdx.x * blockDim.x + threadIdx.x;
    long stride = (long)gridDim.x * blockDim.x;
    for (; i < n; i += stride) bins[i] = 0.0f;
}

// ---------------------------------------------------------------------------
// Kernel 3: scatter-accumulate. One thread per (track, neighbor) pair.
// A 400-sample signal starting at `start` spans <= 9 consecutive 50-wide bins,
// so we reduce each bin's chunk in registers and do one f32 atomic per bin.
// Bins live in a 6.8 MB array -> resident in the 192 MB L2, atomics are cheap.
// ---------------------------------------------------------------------------
__global__ void accumulate_kernel(const int* __restrict__ neigh,
                                  const float* __restrict__ signals,
                                  const int* __restrict__ track_starts,
                                  const int* __restrict__ map,
                                  float* __restrict__ bins,
                                  int NM, int M, int U) {
    int tid = blockIdx.x * blockDim.x + threadIdx.x;
    if (tid >= NM) return;
    int n  = tid / M;
    int px = neigh[2 * tid + 0];
    int py = neigh[2 * tid + 1];
    int pidx = (px < 0) ? U : map[px * GRID_DIM + py];   // dead/miss -> trash row U
    int start = track_starts[n];

    const float* sig = signals + (size_t)tid * T_SIG;
    float* row = bins + (size_t)pidx * NBP;

    int bin = start / ADC_SAMPLE;
    int rem = ADC_SAMPLE - (start - bin * ADC_SAMPLE);
    int t = 0;
    while (t < T_SIG) {
        int c = rem < (T_SIG - t) ? rem : (T_SIG - t);
        float acc = 0.0f;
        for (int i = 0; i < c; ++i) acc += sig[t + i];
        atomicAdd(&row[bin], acc);
        t += c;
        bin += 1;
        rem = ADC_SAMPLE;
    }
}

// ---------------------------------------------------------------------------
// Kernel 4: WMMA prefix-sum scan + ADC.
// One wave handles 16 pixel rows. For each 16-column block b:
//   D = A(16x16 bins) x Tri(16x16, Tri[k][j] = k<=j) + carry
// realized as four chained V_WMMA_F32_16X16X4_F32 accumulations (K=4 each).
// A-layout  (16x4 f32): lanes 0-15 M=lane  K=kb*4+{0,1}; lanes 16-31 K=kb*4+{2,3}
// B-layout  (4x16 f32): VGPR v, lanes 0-15 K=v,  N=lane; lanes 16-31 K=v+2, N=lane-16
// C/D layout(16x16 f32): VGPR v, lanes 0-15 M=v, N=lane; lanes 16-31 M=v+8, N=lane-16
// Carry (last column of D) lives in lanes 15/31 -> broadcast via __shfl and
// seeded into the next block's C accumulator. Guards are wave-uniform, so
// EXEC is all-ones at every WMMA (ISA requirement).
// ---------------------------------------------------------------------------
__global__ void wmma_scan_adc_kernel(const float* __restrict__ bins,
                                     float* __restrict__ out, int U) {
    int wave = (int)((blockIdx.x * blockDim.x + threadIdx.x) >> 5);
    int lane = threadIdx.x & 31;
    int prow = wave * 16;                 // first pixel row of this wave's group
    if (prow >= U) return;                // wave-uniform guard

    int half = lane >> 4;                 // 0: lanes 0-15, 1: lanes 16-31
    int l15  = lane & 15;

    float carry[8];
#pragma unroll
    for (int v = 0; v < 8; ++v) carry[v] = 0.0f;

    const float* base = bins + (size_t)prow * NBP;

    for (int b = 0; b < NBLK; ++b) {
        v8f c;
#pragma unroll
        for (int v = 0; v < 8; ++v) c[v] = carry[v];

        const float* tile = base + b * 16;
#pragma unroll
        for (int kb = 0; kb < 4; ++kb) {
            v2f a, tri;
            int kA = kb * 4 + half * 2;                 // A: this lane's K pair
            a.x = tile[(size_t)l15 * NBP + kA + 0];
            a.y = tile[(size_t)l15 * NBP + kA + 1];
            int kB = kb * 4 + half * 2;                 // B: triangular ones
            tri.x = ((kB + 0) <= l15) ? 1.0f : 0.0f;
            tri.y = ((kB + 1) <= l15) ? 1.0f : 0.0f;
            c = __builtin_amdgcn_wmma_f32_16x16x4_f32(
                    /*neg_a=*/false, a, /*neg_b=*/false, tri,
                    /*c_mod=*/(short)0, c, /*reuse_a=*/false, /*reuse_b=*/false);
        }

        // Emit ADC for valid columns of this block.
        int colbase = b * 16;
#pragma unroll
        for (int v = 0; v < 8; ++v) {
            int Mrow = v + half * 8;
            int col  = colbase + l15;
            if (col < NB) {
                float val = c[v] * 0.25f + 74.0f;       // GAIN, PEDESTAL
                val = fminf(fmaxf(val, 0.0f), 255.0f);  // clip to ADC_MAX
                out[(size_t)(prow + Mrow) * NB + col] = val;
            }
        }

        // Running carry: last column (N=15) sits in lane 15 (rows v) / 31 (rows v+8).
        int src = half ? 31 : 15;
#pragma unroll
        for (int v = 0; v < 8; ++v) carry[v] = __shfl(c[v], src, 32);
    }
}

// ---------------------------------------------------------------------------
// Launcher
// ---------------------------------------------------------------------------
extern "C" void kernel_launch(void* const* d_in, const int* in_sizes, int n_in,
                              void* d_out, int out_size, void* d_ws, size_t ws_size,
                              hipStream_t stream) {
    const int*   neigh   = (const int*)d_in[0];   // (N, M, 2) int32
    const int*   pool    = (const int*)d_in[1];   // (U, 2) int32
    const float* signals = (const float*)d_in[2]; // (N, M, T_SIG) f32
    const int*   starts  = (const int*)d_in[3];   // (N,) int32
    float*       out     = (float*)d_out;         // (U, NB) f32

    int U  = in_sizes[1] / 2;
    int N  = in_sizes[3];
    int NM = in_sizes[0] / 2;
    int M  = NM / N;

    // Workspace layout: [ map: 128*128 int32 | bins: (U+1)*NBP f32 ]
    int*   map  = (int*)d_ws;
    float* bins = (float*)((char*)d_ws + (size_t)GRID_DIM * GRID_DIM * sizeof(int));
    long   nbins = (long)(U + 1) * NBP;

    init_map_kernel<<<(GRID_DIM * GRID_DIM + 255) / 256, 256, 0, stream>>>(map, U);
    fill_map_kernel<<<(U + 255) / 256, 256, 0, stream>>>(pool, map, U);
    zero_bins_kernel<<<256, 256, 0, stream>>>(bins, nbins);
    accumulate_kernel<<<(NM + 127) / 128, 128, 0, stream>>>(
        neigh, signals, starts, map, bins, NM, M, U);

    int waves   = (U + 15) / 16;               // one wave per 16 pixel rows
    int threads = waves * 32;
    wmma_scan_adc_kernel<<<(threads + 255) / 256, 256, 0, stream>>>(bins, out, U);
}